// IsoMaxLossFirstPart_73031623901227
// MI455X (gfx1250) — compile-verified
//
#include <hip/hip_runtime.h>

// ---------------------------------------------------------------------------
// IsoMax first part: logits[b,c] = -|| (features[b]+EPS) - prototypes[c] ||_2
//   ||x+e - p||^2 = ||x+e||^2 + ( ||p||^2 - 2e*sum(p) ) - 2 * (x . p)
// Cross term on V_WMMA_F32_16X16X4_F32 (native f32 matrix path, wave32).
// Double-buffered LDS; async global->LDS (ASYNCcnt) when toolchain exposes it.
// B=16384, F=2048, C=1000 (from reference setup_inputs()).
// ---------------------------------------------------------------------------

#define EPS_VAL 1e-6f
#define B_DIM 16384
#define F_DIM 2048
#define C_DIM 1000

#define BM 128
#define BN 128
#define BK 32
#define LDK 36                    // padded LDS row stride (floats); 144B rows
#define TILE_FLOATS (128 * LDK)   // one 128x32 staged tile (padded)

typedef __attribute__((ext_vector_type(2))) float v2f;
typedef __attribute__((ext_vector_type(4))) float v4f;
typedef __attribute__((ext_vector_type(8))) float v8f;

#if defined(__gfx1250__) && \
    __has_builtin(__builtin_amdgcn_global_load_async_to_lds_b128) && \
    __has_builtin(__builtin_amdgcn_s_wait_asynccnt)
#define USE_ASYNC 1
typedef int v4i_t __attribute__((vector_size(16)));
typedef __attribute__((address_space(1))) v4i_t* as1_v4i;   // global int4*
typedef __attribute__((address_space(3))) v4i_t* as3_v4i;   // LDS    int4*
#else
#define USE_ASYNC 0
#endif

// -------- xsq[row] = sum_k (features[row,k] + EPS)^2 ------------------------
__global__ __launch_bounds__(256)
void feat_sq_kernel(const float* __restrict__ in, float* __restrict__ out,
                    int ncols) {
  __shared__ float red[256];
  const int row = blockIdx.x;
  const float* p = in + (size_t)row * ncols;
  float s = 0.0f;
  for (int i = threadIdx.x; i < ncols; i += 256) {
    float v = p[i] + EPS_VAL;
    s = fmaf(v, v, s);
  }
  red[threadIdx.x] = s;
  __syncthreads();
  #pragma unroll
  for (int off = 128; off > 0; off >>= 1) {
    if (threadIdx.x < off) red[threadIdx.x] += red[threadIdx.x + off];
    __syncthreads();
  }
  if (threadIdx.x == 0) out[row] = red[0];
}

// -------- adj[row] = ||p_row||^2 - 2*EPS*sum(p_row) -------------------------
__global__ __launch_bounds__(256)
void proto_adj_kernel(const float* __restrict__ in, float* __restrict__ adj,
                      int ncols) {
  __shared__ float r_sq[256];
  __shared__ float r_s[256];
  const int row = blockIdx.x;
  const float* p = in + (size_t)row * ncols;
  float ssq = 0.0f, s = 0.0f;
  for (int i = threadIdx.x; i < ncols; i += 256) {
    float v = p[i];
    ssq = fmaf(v, v, ssq);
    s += v;
  }
  r_sq[threadIdx.x] = ssq;
  r_s[threadIdx.x]  = s;
  __syncthreads();
  #pragma unroll
  for (int off = 128; off > 0; off >>= 1) {
    if (threadIdx.x < off) {
      r_sq[threadIdx.x] += r_sq[threadIdx.x + off];
      r_s[threadIdx.x]  += r_s[threadIdx.x + off];
    }
    __syncthreads();
  }
  if (threadIdx.x == 0) adj[row] = r_sq[0] - 2.0f * EPS_VAL * r_s[0];
}

// -------- stage one 128 x BK f32 tile into LDS ------------------------------
// g points at (tile_row0, tile_k0); 4 x b128 per thread covers 128x32 floats.
__device__ __forceinline__
void stage_tile128(const float* __restrict__ g, int rows_valid,
                   float* lds_base, int tid) {
  #pragma unroll
  for (int i = 0; i < 4; ++i) {
    const int idx = tid + i * 256;        // 0..1023
    const int row = idx >> 3;             // 0..127
    const int c4  = (idx & 7) << 2;       // 0,4,...,28
    float* dst = lds_base + row * LDK + c4;
    if (row < rows_valid) {
#if USE_ASYNC
      __builtin_amdgcn_global_load_async_to_lds_b128(
          (as1_v4i)(g + (size_t)row * F_DIM + c4),
          (as3_v4i)dst, 0, 0);
#else
      __builtin_prefetch(g + (size_t)row * F_DIM + c4 + BK, 0, 0);
      *(v4f*)dst = *(const v4f*)(g + (size_t)row * F_DIM + c4);
#endif
    } else {
      *(v4f*)dst = (v4f){};               // pad rows beyond C with zeros
    }
  }
}

// -------- fused cross-GEMM + distance epilogue ------------------------------
__global__ __launch_bounds__(256)
void dist_gemm_kernel(const float* __restrict__ A,    // features  (B,F)
                      const float* __restrict__ P,    // prototypes(C,F)
                      const float* __restrict__ xsq,  // (B)
                      const float* __restrict__ padj, // (C)
                      float* __restrict__ out) {      // (B,C)
  __shared__ float As[2][TILE_FLOATS];
  __shared__ float Bs[2][TILE_FLOATS];

  const int tid  = threadIdx.x;
  const int lane = tid & 31;
  const int wave = tid >> 5;     // 8 waves
  const int wm   = wave >> 1;    // 0..3 -> 32-row patch
  const int wn   = wave & 1;     // 0..1 -> 64-col patch
  const int m0   = blockIdx.y * BM;
  const int n0   = blockIdx.x * BN;
  const int brows = (C_DIM - n0 < BN) ? (C_DIM - n0) : BN;

  const int half = lane >> 4;    // lanes 0-15 vs 16-31
  const int ln   = lane & 15;
  const int kb   = half * 2;     // K-pair base per f32 WMMA A/B layout

  v8f acc[2][4];
  #pragma unroll
  for (int tm = 0; tm < 2; ++tm)
    #pragma unroll
    for (int tn = 0; tn < 4; ++tn) acc[tm][tn] = (v8f){};

  const float* gA = A + (size_t)m0 * F_DIM;
  const float* gB = P + (size_t)n0 * F_DIM;

  // prologue: stage tile 0 into buffer 0
  stage_tile128(gA, 128, As[0], tid);
  stage_tile128(gB, brows, Bs[0], tid);

  const int ntiles = F_DIM / BK;           // 64
  for (int t = 0; t < ntiles; ++t) {
    const int cur = t & 1;
#if USE_ASYNC
    __builtin_amdgcn_s_wait_asynccnt(0);   // my async writes landed in LDS
#endif
    __syncthreads();                       // everyone's tile data visible

    if (t + 1 < ntiles) {                  // refill the other buffer
      const int kt = (t + 1) * BK;
      stage_tile128(gA + kt, 128, As[cur ^ 1], tid);
      stage_tile128(gB + kt, brows, Bs[cur ^ 1], tid);
    }

    // 8 WMMA k-steps (K=4) over BK=32; 6 b64 LDS frag loads -> 8 WMMAs
    #pragma unroll
    for (int k0 = 0; k0 < BK; k0 += 4) {
      v2f a[2], b[4];
      #pragma unroll
      for (int tm = 0; tm < 2; ++tm)
        a[tm] = *(const v2f*)&As[cur][(wm * 32 + tm * 16 + ln) * LDK + k0 + kb];
      #pragma unroll
      for (int tn = 0; tn < 4; ++tn)
        b[tn] = *(const v2f*)&Bs[cur][(wn * 64 + tn * 16 + ln) * LDK + k0 + kb];
      #pragma unroll
      for (int tm = 0; tm < 2; ++tm)
        #pragma unroll
        for (int tn = 0; tn < 4; ++tn)
          acc[tm][tn] = __builtin_amdgcn_wmma_f32_16x16x4_f32(
              false, a[tm], false, b[tn], (short)0, acc[tm][tn], false, false);
    }
    __syncthreads();                       // done reading buf[cur] before refill
  }

  // epilogue: out = -sqrt(max(xsq[m] + adj[n] - 2*cross, 0))
  // C/D layout: lane ln -> N, VGPR r -> M = r + 8*half
  const int om_base = m0 + wm * 32;
  const int on_base = n0 + wn * 64;
  #pragma unroll
  for (int tm = 0; tm < 2; ++tm) {
    #pragma unroll
    for (int tn = 0; tn < 4; ++tn) {
      const int n = on_base + tn * 16 + ln;
      if (n < C_DIM) {
        const float an = padj[n];
        #pragma unroll
        for (int r = 0; r < 8; ++r) {
          const int m = om_base + tm * 16 + half * 8 + r;
          float sq = xsq[m] + an - 2.0f * acc[tm][tn][r];
          sq = fmaxf(sq, 0.0f);
          out[(size_t)m * C_DIM + n] = -__builtin_sqrtf(sq);
        }
      }
    }
  }
}

extern "C" void kernel_launch(void* const* d_in, const int* in_sizes, int n_in,
                              void* d_out, int out_size, void* d_ws, size_t ws_size,
                              hipStream_t stream) {
  (void)in_sizes; (void)n_in; (void)out_size; (void)ws_size;
  const float* features   = (const float*)d_in[0]; // (16384, 2048) f32
  const float* prototypes = (const float*)d_in[1]; // (1000, 2048)  f32
  float* out = (float*)d_out;                      // (16384, 1000) f32

  float* xsq  = (float*)d_ws;          // 16384 floats
  float* padj = xsq + B_DIM;           // 1000 floats

  feat_sq_kernel<<<B_DIM, 256, 0, stream>>>(features, xsq, F_DIM);
  proto_adj_kernel<<<C_DIM, 256, 0, stream>>>(prototypes, padj, F_DIM);

  dim3 grid((C_DIM + BN - 1) / BN, B_DIM / BM);    // (8, 128)
  dist_gemm_kernel<<<grid, 256, 0, stream>>>(features, prototypes, xsq, padj, out);
}